// SNNLinear_23407571763420
// MI455X (gfx1250) — compile-verified
//
#include <hip/hip_runtime.h>

typedef __attribute__((ext_vector_type(2))) float v2f;
typedef __attribute__((ext_vector_type(8))) float v8f;

#define BROWS  8192
#define D_IN   784
#define D_OUT  512
#define STEPS  25
#define THRESH 1.0f
#define BETA   0.95f

// One wave computes a 16x16 tile of cur = x @ W^T + b via f32 WMMA,
// then runs the 25-step LIF recurrence entirely in registers,
// streaming spk/mem to global with non-temporal stores.
__global__ __launch_bounds__(256, 2)
void snn_fused_kernel(const float* __restrict__ x,
                      const float* __restrict__ W,
                      const float* __restrict__ bias,
                      float* __restrict__ out)
{
    const int lane = threadIdx.x & 31;
    const int wave = threadIdx.x >> 5;

    const int col_tile = blockIdx.x * 8 + wave;   // 32 column tiles (512/16)
    const int row_tile = blockIdx.y;              // 512 row tiles (8192/16)
    const int row0 = row_tile * 16;
    const int col0 = col_tile * 16;

    const int half = lane >> 4;   // 0: lanes 0-15, 1: lanes 16-31
    const int l15  = lane & 15;

    // A fragment (16x4 f32): lane l15 = row M, VGPR pair holds K = {2*half, 2*half+1}
    const float* xrow = x + (size_t)(row0 + l15) * D_IN + 2 * half;
    // B fragment (4x16 f32): B[k][n] = W[n][k]; lane l15 = col N, K = {2*half, 2*half+1}
    const float* wrow = W + (size_t)(col0 + l15) * D_IN + 2 * half;

    v8f c = {};
    #pragma unroll 4
    for (int k = 0; k < D_IN; k += 4) {
        v2f a = *(const v2f*)(xrow + k);
        v2f b = *(const v2f*)(wrow + k);
        // D = A(16x4) * B(4x16) + C, exact f32
        c = __builtin_amdgcn_wmma_f32_16x16x4_f32(
                /*neg_a=*/false, a, /*neg_b=*/false, b,
                /*c_mod=*/(short)0, c, /*reuse_a=*/false, /*reuse_b=*/false);
    }

    // Every accumulator VGPR in this lane has N = l15 -> single bias value.
    const float bv = bias[col0 + l15];
    #pragma unroll
    for (int r = 0; r < 8; ++r) c[r] += bv;

    // LIF dynamics: mem starts at 0; reset uses PREVIOUS mem (subtract reset).
    float mem[8];
    #pragma unroll
    for (int r = 0; r < 8; ++r) mem[r] = 0.0f;

    const size_t plane   = (size_t)BROWS * D_OUT;       // 8192*512
    const size_t mem_off = (size_t)STEPS * plane;       // start of mem_rec
    // VGPR r maps to element (row0 + r + 8*half, col0 + l15)
    const size_t base = (size_t)(row0 + 8 * half) * D_OUT + (size_t)(col0 + l15);

    for (int t = 0; t < STEPS; ++t) {
        float* spk_t = out + (size_t)t * plane + base;
        float* mem_t = out + mem_off + (size_t)t * plane + base;
        #pragma unroll
        for (int r = 0; r < 8; ++r) {
            const float reset = (mem[r] > THRESH) ? THRESH : 0.0f;
            const float m = __builtin_fmaf(BETA, mem[r], c[r]) - reset;
            mem[r] = m;
            const float spk = (m > THRESH) ? 1.0f : 0.0f;
            __builtin_nontemporal_store(spk, spk_t + (size_t)r * D_OUT);
            __builtin_nontemporal_store(m,   mem_t + (size_t)r * D_OUT);
        }
    }
}

extern "C" void kernel_launch(void* const* d_in, const int* in_sizes, int n_in,
                              void* d_out, int out_size, void* d_ws, size_t ws_size,
                              hipStream_t stream) {
    (void)in_sizes; (void)n_in; (void)d_ws; (void)ws_size; (void)out_size;
    const float* x    = (const float*)d_in[0];   // [8192, 784]
    const float* W    = (const float*)d_in[1];   // [512, 784]
    const float* bias = (const float*)d_in[2];   // [512]
    // d_in[3] = rate_encoding (int scalar), unused (reference takes the static path)
    float* out = (float*)d_out;                  // [25,8192,512] spk ++ [25,8192,512] mem

    dim3 grid(4, 512);   // 4 blocks * 8 waves = 32 col tiles; 512 row tiles
    dim3 block(256);
    snn_fused_kernel<<<grid, block, 0, stream>>>(x, W, bias, out);
}